// GraphTripleConv_3530463117740
// MI455X (gfx1250) — compile-verified
//
#include <hip/hip_runtime.h>

#define O_N   50000
#define T_N   200000
#define DIN   128
#define H_N   512
#define DOUT  128
#define W1A_K 384      // 3*DIN
#define W1B_N 1152     // 2*H + DOUT

typedef float v2f __attribute__((ext_vector_type(2)));
typedef float v8f __attribute__((ext_vector_type(8)));

__device__ __forceinline__ v8f wmma_f32(v2f a, v2f b, v8f c) {
  // D = A(16x4 f32) * B(4x16 f32) + C(16x16 f32), fp32 matrix pipe
  return __builtin_amdgcn_wmma_f32_16x16x4_f32(false, a, false, b, (short)0, c,
                                               false, false);
}

__global__ void zero_kernel(float* __restrict__ p, long n) {
  long i = (long)blockIdx.x * blockDim.x + threadIdx.x;
  long stride = (long)gridDim.x * blockDim.x;
  for (; i < n; i += stride) p[i] = 0.0f;
}

// One workgroup (8 wave32) per 32-triple tile (2 WMMA M-tiles share every
// B-operand load -> 2x fewer L2 weight bytes per FLOP).
// Fuses: gather -> GEMM1(384x512)+ReLU -> GEMM2(512x1152)+ReLU -> route.
__global__ __launch_bounds__(256) void triple_kernel(
    const float* __restrict__ obj, const float* __restrict__ pred,
    const int* __restrict__ edges,
    const float* __restrict__ W1a, const float* __restrict__ b1a,
    const float* __restrict__ W1b, const float* __restrict__ b1b,
    float* __restrict__ out_newp, float* __restrict__ pooled,
    float* __restrict__ counts)
{
  __shared__ float sIn[32 * W1A_K];   // 48 KB: cur_t tile, stride 384
  __shared__ float sH[32 * H_N];      // 64 KB: h tile, stride 512
  __shared__ int sS[32];
  __shared__ int sO[32];

  const int tid   = threadIdx.x;
  const int wave  = tid >> 5;
  const int lane  = tid & 31;
  const int lo16  = lane & 15;
  const int khalf = (lane >> 4) << 1;   // lanes 0-15 -> K=0,1 ; lanes 16-31 -> K=2,3
  const int mhalf = (lane >> 4) << 3;   // C/D: vgpr r -> row r (+8 for upper half-wave)
  const int t0    = blockIdx.x * 32;

  if (tid < 32) {
    sS[tid] = edges[(t0 + tid) * 2 + 0];
    sO[tid] = edges[(t0 + tid) * 2 + 1];
  }
  __syncthreads();

  // Gather cur_t = [obj[s] | pred | obj[o]] : 32 x 384
  for (int i = tid; i < 32 * W1A_K; i += 256) {
    int r = i / W1A_K, c = i - r * W1A_K;
    float v;
    if (c < DIN)          v = obj[(long)sS[r] * DIN + c];
    else if (c < 2 * DIN) v = pred[(long)(t0 + r) * DIN + (c - DIN)];
    else                  v = obj[(long)sO[r] * DIN + (c - 2 * DIN)];
    sIn[r * W1A_K + c] = v;
  }
  __syncthreads();

  // ---- GEMM1: h = relu(cur_t @ W1a + b1a); 32 N-chunks, 4 per wave, 2 M-tiles ----
  for (int j = 0; j < 4; j++) {
    const int n = (wave * 4 + j) * 16 + lo16;
    const float bias = b1a[n];
    v8f acc0, acc1;
    #pragma unroll
    for (int r = 0; r < 8; r++) { acc0[r] = bias; acc1[r] = bias; }
    for (int kk = 0; kk < W1A_K / 4; kk++) {
      const int k = kk * 4 + khalf;
      v2f b, a0, a1;
      b.x  = W1a[(long)k * H_N + n];
      b.y  = W1a[(long)(k + 1) * H_N + n];
      a0.x = sIn[lo16 * W1A_K + k];
      a0.y = sIn[lo16 * W1A_K + k + 1];
      a1.x = sIn[(16 + lo16) * W1A_K + k];
      a1.y = sIn[(16 + lo16) * W1A_K + k + 1];
      acc0 = wmma_f32(a0, b, acc0);
      acc1 = wmma_f32(a1, b, acc1);
    }
    #pragma unroll
    for (int r = 0; r < 8; r++) {
      float v0 = acc0[r], v1 = acc1[r];
      sH[(mhalf + r) * H_N + n]        = v0 > 0.0f ? v0 : 0.0f;
      sH[(16 + mhalf + r) * H_N + n]   = v1 > 0.0f ? v1 : 0.0f;
    }
  }
  __syncthreads();

  // ---- GEMM2: new_t = relu(h @ W1b + b1b); 72 N-chunks, 9 per wave, 2 M-tiles ----
  for (int cc = wave; cc < W1B_N / 16; cc += 8) {
    const int n = cc * 16 + lo16;
    const float bias = b1b[n];
    v8f acc[2];
    #pragma unroll
    for (int r = 0; r < 8; r++) { acc[0][r] = bias; acc[1][r] = bias; }
    for (int kk = 0; kk < H_N / 4; kk++) {
      const int k = kk * 4 + khalf;
      if ((kk & 31) == 0 && kk < 96)
        __builtin_prefetch(&W1b[(long)(k + 64) * W1B_N + n], 0, 0);
      v2f b, a0, a1;
      b.x  = W1b[(long)k * W1B_N + n];
      b.y  = W1b[(long)(k + 1) * W1B_N + n];
      a0.x = sH[lo16 * H_N + k];
      a0.y = sH[lo16 * H_N + k + 1];
      a1.x = sH[(16 + lo16) * H_N + k];
      a1.y = sH[(16 + lo16) * H_N + k + 1];
      acc[0] = wmma_f32(a0, b, acc[0]);
      acc[1] = wmma_f32(a1, b, acc[1]);
    }
    // Route: cols [0,512) -> new_s scatter, [512,640) -> new_p, [640,1152) -> new_o
    #pragma unroll
    for (int mt = 0; mt < 2; mt++) {
      #pragma unroll
      for (int r = 0; r < 8; r++) {
        const int m = mt * 16 + mhalf + r;
        float v = acc[mt][r] > 0.0f ? acc[mt][r] : 0.0f;
        if (n < H_N) {
          atomicAdd(&pooled[(long)sS[m] * H_N + n], v);
        } else if (n < H_N + DOUT) {
          out_newp[(long)(t0 + m) * DOUT + (n - H_N)] = v;
        } else {
          atomicAdd(&pooled[(long)sO[m] * H_N + (n - H_N - DOUT)], v);
        }
      }
    }
  }

  if (tid < 32) {
    atomicAdd(&counts[sS[tid]], 1.0f);
    atomicAdd(&counts[sO[tid]], 1.0f);
  }
}

// One workgroup per 32-object tile (rows guarded; O % 32 != 0): normalize
// pooled, GEMM3(512x512)+ReLU, GEMM4(512x128)+ReLU -> new_obj.
__global__ __launch_bounds__(256) void obj_kernel(
    const float* __restrict__ pooled, const float* __restrict__ counts,
    const float* __restrict__ W2a, const float* __restrict__ b2a,
    const float* __restrict__ W2b, const float* __restrict__ b2b,
    float* __restrict__ out_obj)
{
  __shared__ float sBuf[32 * H_N];   // 64 KB, reused: normalized pooled then h2
  __shared__ float sInv[32];

  const int tid   = threadIdx.x;
  const int wave  = tid >> 5;
  const int lane  = tid & 31;
  const int lo16  = lane & 15;
  const int khalf = (lane >> 4) << 1;
  const int mhalf = (lane >> 4) << 3;
  const int o0    = blockIdx.x * 32;

  if (tid < 32) {
    const int oi = o0 + tid;
    float c = (oi < O_N) ? counts[oi] : 1.0f;
    c = c < 1.0f ? 1.0f : c;
    c = c > (float)O_N ? (float)O_N : c;
    sInv[tid] = 1.0f / c;
  }
  __syncthreads();
  for (int i = tid; i < 32 * H_N; i += 256) {
    const int r = i >> 9;
    const int oi = o0 + r;
    sBuf[i] = (oi < O_N) ? pooled[(long)oi * H_N + (i & (H_N - 1))] * sInv[r]
                         : 0.0f;
  }
  __syncthreads();

  // ---- GEMM3: h2 = relu(pooled_n @ W2a + b2a); 4 N-chunks/wave, 2 M-tiles,
  //      accumulators held in VGPRs across the barrier so sBuf can be reused ----
  v8f accA[4], accB[4];
  #pragma unroll
  for (int j = 0; j < 4; j++) {
    const int n = (wave * 4 + j) * 16 + lo16;
    const float bias = b2a[n];
    #pragma unroll
    for (int r = 0; r < 8; r++) { accA[j][r] = bias; accB[j][r] = bias; }
    for (int kk = 0; kk < H_N / 4; kk++) {
      const int k = kk * 4 + khalf;
      v2f b, a0, a1;
      b.x  = W2a[(long)k * H_N + n];
      b.y  = W2a[(long)(k + 1) * H_N + n];
      a0.x = sBuf[lo16 * H_N + k];
      a0.y = sBuf[lo16 * H_N + k + 1];
      a1.x = sBuf[(16 + lo16) * H_N + k];
      a1.y = sBuf[(16 + lo16) * H_N + k + 1];
      accA[j] = wmma_f32(a0, b, accA[j]);
      accB[j] = wmma_f32(a1, b, accB[j]);
    }
  }
  __syncthreads();                      // everyone done reading normalized pooled
  #pragma unroll
  for (int j = 0; j < 4; j++) {
    const int n = (wave * 4 + j) * 16 + lo16;
    #pragma unroll
    for (int r = 0; r < 8; r++) {
      float v0 = accA[j][r], v1 = accB[j][r];
      sBuf[(mhalf + r) * H_N + n]      = v0 > 0.0f ? v0 : 0.0f;
      sBuf[(16 + mhalf + r) * H_N + n] = v1 > 0.0f ? v1 : 0.0f;
    }
  }
  __syncthreads();

  // ---- GEMM4: new_obj = relu(h2 @ W2b + b2b); N=128 -> 1 chunk/wave, 2 M-tiles ----
  {
    const int n = wave * 16 + lo16;
    const float bias = b2b[n];
    v8f acc0, acc1;
    #pragma unroll
    for (int r = 0; r < 8; r++) { acc0[r] = bias; acc1[r] = bias; }
    for (int kk = 0; kk < H_N / 4; kk++) {
      const int k = kk * 4 + khalf;
      v2f b, a0, a1;
      b.x  = W2b[(long)k * DOUT + n];
      b.y  = W2b[(long)(k + 1) * DOUT + n];
      a0.x = sBuf[lo16 * H_N + k];
      a0.y = sBuf[lo16 * H_N + k + 1];
      a1.x = sBuf[(16 + lo16) * H_N + k];
      a1.y = sBuf[(16 + lo16) * H_N + k + 1];
      acc0 = wmma_f32(a0, b, acc0);
      acc1 = wmma_f32(a1, b, acc1);
    }
    #pragma unroll
    for (int r = 0; r < 8; r++) {
      const int row0 = o0 + mhalf + r;
      const int row1 = o0 + 16 + mhalf + r;
      float v0 = acc0[r] > 0.0f ? acc0[r] : 0.0f;
      float v1 = acc1[r] > 0.0f ? acc1[r] : 0.0f;
      if (row0 < O_N) out_obj[(long)row0 * DOUT + n] = v0;
      if (row1 < O_N) out_obj[(long)row1 * DOUT + n] = v1;
    }
  }
}

extern "C" void kernel_launch(void* const* d_in, const int* in_sizes, int n_in,
                              void* d_out, int out_size, void* d_ws, size_t ws_size,
                              hipStream_t stream) {
  const float* obj   = (const float*)d_in[0];
  const float* pred  = (const float*)d_in[1];
  const int*   edges = (const int*)d_in[2];
  const float* W1a   = (const float*)d_in[3];
  const float* b1a   = (const float*)d_in[4];
  const float* W1b   = (const float*)d_in[5];
  const float* b1b   = (const float*)d_in[6];
  const float* W2a   = (const float*)d_in[7];
  const float* b2a   = (const float*)d_in[8];
  const float* W2b   = (const float*)d_in[9];
  const float* b2b   = (const float*)d_in[10];

  float* out_obj  = (float*)d_out;                     // new_obj: O x 128
  float* out_newp = out_obj + (long)O_N * DOUT;        // new_p:   T x 128

  float* pooled = (float*)d_ws;                        // O x 512
  float* counts = pooled + (long)O_N * H_N;            // O

  const long nz = (long)O_N * H_N + O_N;
  zero_kernel<<<dim3((unsigned)((nz + 255) / 256)), dim3(256), 0, stream>>>(pooled, nz);
  triple_kernel<<<dim3(T_N / 32), dim3(256), 0, stream>>>(
      obj, pred, edges, W1a, b1a, W1b, b1b, out_newp, pooled, counts);
  obj_kernel<<<dim3((O_N + 31) / 32), dim3(256), 0, stream>>>(
      pooled, counts, W2a, b2a, W2b, b2b, out_obj);
}